// SKMAttention_83227876262483
// MI455X (gfx1250) — compile-verified
//
#include <hip/hip_runtime.h>
#include <hip/hip_bf16.h>
#include <math.h>

// ---------------- problem constants ----------------
constexpr int D_MODEL  = 96;
constexpr int D_STATE  = 16;
constexpr int D_INNER  = 192;
constexpr int DT_RANK  = 6;
constexpr int KDIR     = 4;
constexpr int Bsz      = 2;
constexpr int Hh       = 48;
constexpr int Ww       = 48;
constexpr int L        = Hh * Ww;          // 2304
constexpr int ML       = Bsz * L;          // 4608 rows for (b,h,w)-major GEMMs
constexpr int NPROJ    = DT_RANK + 2 * D_STATE; // 38
constexpr int NPROJ_P  = 48;               // padded to 3 wmma tiles

typedef float v2f __attribute__((ext_vector_type(2)));
typedef float v8f __attribute__((ext_vector_type(8)));

// ---------------------------------------------------------------------------
// One 16x16 C tile of  C = A[m0:m0+16, 0:Kd] * W[n0:n0+16, 0:Kd]^T   (fp32)
// using V_WMMA_F32_16X16X4_F32.  A: row-major lda.  W: row-major ldw (N x K).
// Rows n >= Nvalid of W are treated as zero (N padding).
// Fragment layouts per CDNA5 ISA 7.12.2:
//   A 16x4 : lanes 0-15 -> M=lane, vgpr{0,1}=K{0,1};  lanes 16-31 -> K{2,3}
//   B 4x16 : lanes 0-15 -> N=lane, vgpr{0,1}=K{0,1};  lanes 16-31 -> K{2,3}
//   C 16x16: vgpr v: lanes 0-15 -> M=v,N=lane; lanes 16-31 -> M=v+8,N=lane-16
// ---------------------------------------------------------------------------
__device__ inline v8f wmma_tile_f32(const float* __restrict__ A, int lda,
                                    const float* __restrict__ W, int ldw,
                                    int m0, int n0, int Kd, int Nvalid)
{
    const int lane = threadIdx.x & 31;
    const int half = lane >> 4;   // 0: K 0/1, 1: K 2/3
    const int lr   = lane & 15;

    v8f c = {0.f, 0.f, 0.f, 0.f, 0.f, 0.f, 0.f, 0.f};
    const int n = n0 + lr;
    const float* arow = A + (size_t)(m0 + lr) * lda + half * 2;
    const float* wrow = W + (size_t)n * ldw + half * 2;
    const bool nok = (n < Nvalid);

    for (int kk = 0; kk < Kd; kk += 4) {
        v2f a, b;
        a.x = arow[kk + 0];
        a.y = arow[kk + 1];
        b.x = nok ? wrow[kk + 0] : 0.f;
        b.y = nok ? wrow[kk + 1] : 0.f;
        // (neg_a, A, neg_b, B, c_mod, C, reuse_a, reuse_b)
        c = __builtin_amdgcn_wmma_f32_16x16x4_f32(false, a, false, b,
                                                  (short)0, c, false, false);
    }
    return c;
}

__device__ inline float silu_f(float v) { return v / (1.f + expf(-v)); }

// ---------------------------------------------------------------------------
// K1: xz = x @ in_proj_w^T ; split into xc (first 192) and z (last 192)
// M=4608, N=384, K=96.  One wave per 16x16 tile, 8 waves/block.
// ---------------------------------------------------------------------------
__global__ void k_in_proj(const float* __restrict__ x,
                          const float* __restrict__ Wp,
                          float* __restrict__ xc, float* __restrict__ z)
{
    constexpr int NT = 2 * D_INNER / 16;           // 24 tiles in N
    const int tile = blockIdx.x * (blockDim.x >> 5) + (threadIdx.x >> 5);
    const int mt = tile / NT, nt = tile % NT;

    v8f c = wmma_tile_f32(x, D_MODEL, Wp, D_MODEL, mt * 16, nt * 16,
                          D_MODEL, 2 * D_INNER);

    const int lane = threadIdx.x & 31;
    const int half = lane >> 4, lr = lane & 15;
    const int n = nt * 16 + lr;
#pragma unroll
    for (int v = 0; v < 8; ++v) {
        const int m = mt * 16 + v + half * 8;
        const float val = c[v];
        if (n < D_INNER) xc[(size_t)m * D_INNER + n] = val;
        else             z [(size_t)m * D_INNER + (n - D_INNER)] = val;
    }
}

// ---------------------------------------------------------------------------
// K2: depthwise 3x3 SAME conv + bias + SiLU; scatter into 4 direction layouts
// xs[b,k,l,d]:  k=0 row-major, k=1 transposed, k=2/3 flipped copies.
// ---------------------------------------------------------------------------
__global__ void k_conv_dirs(const float* __restrict__ xc,
                            const float* __restrict__ cw,
                            const float* __restrict__ cb,
                            float* __restrict__ xs)
{
    const int idx = blockIdx.x * blockDim.x + threadIdx.x;
    if (idx >= Bsz * L * D_INNER) return;
    const int d = idx % D_INNER;
    int t = idx / D_INNER;
    const int w = t % Ww; t /= Ww;
    const int h = t % Hh;
    const int b = t / Hh;

    float acc = cb[d];
#pragma unroll
    for (int kh = 0; kh < 3; ++kh) {
#pragma unroll
        for (int kw = 0; kw < 3; ++kw) {
            const int hh = h + kh - 1, ww = w + kw - 1;
            if (hh >= 0 && hh < Hh && ww >= 0 && ww < Ww)
                acc += xc[((size_t)(b * Hh + hh) * Ww + ww) * D_INNER + d] *
                       cw[d * 9 + kh * 3 + kw];
        }
    }
    const float v = silu_f(acc);

    const size_t base = (size_t)b * KDIR * L * D_INNER;
    const int l  = h * Ww + w;
    const int lt = w * Hh + h;
    xs[base + ((size_t)0 * L + l)            * D_INNER + d] = v;
    xs[base + ((size_t)1 * L + lt)           * D_INNER + d] = v;
    xs[base + ((size_t)2 * L + (L - 1 - l))  * D_INNER + d] = v;
    xs[base + ((size_t)3 * L + (L - 1 - lt)) * D_INNER + d] = v;
}

// ---------------------------------------------------------------------------
// K3: x_dbl[b,k,l,c] = sum_d xs[b,k,l,d] * x_proj_weight[k,c,d]
// Per (b,k): M=2304, N=48 (valid 38), K=192.  Stored row stride NPROJ_P.
// ---------------------------------------------------------------------------
__global__ void k_xdbl(const float* __restrict__ xs,
                       const float* __restrict__ xpw,
                       float* __restrict__ xdbl)
{
    constexpr int TM = L / 16;                     // 144
    constexpr int TN = NPROJ_P / 16;               // 3
    constexpr int TPK = TM * TN;                   // 432 tiles per (b,k)
    const int tile = blockIdx.x * (blockDim.x >> 5) + (threadIdx.x >> 5);
    const int bk = tile / TPK;
    const int t  = tile % TPK;
    const int mt = t / TN, nt = t % TN;
    const int k  = bk % KDIR;

    const float* A  = xs  + (size_t)bk * L * D_INNER;
    const float* Wk = xpw + (size_t)k * NPROJ * D_INNER;
    v8f c = wmma_tile_f32(A, D_INNER, Wk, D_INNER, mt * 16, nt * 16,
                          D_INNER, NPROJ);

    float* out = xdbl + (size_t)bk * L * NPROJ_P;
    const int lane = threadIdx.x & 31;
    const int half = lane >> 4, lr = lane & 15;
    const int n = nt * 16 + lr;
#pragma unroll
    for (int v = 0; v < 8; ++v) {
        const int m = mt * 16 + v + half * 8;
        out[(size_t)m * NPROJ_P + n] = c[v];
    }
}

// ---------------------------------------------------------------------------
// K4: selective scan.  Block per (b,k), thread per d (192 threads).
// delta = softplus(dts @ Wdt + bias);  h_n <- exp(delta*A_n) h_n + delta*u*B_n
// y = sum_n h_n * C_n + D*u.   Sequential over L in the direction's order.
// ---------------------------------------------------------------------------
__global__ void k_scan(const float* __restrict__ xs,
                       const float* __restrict__ xdbl,
                       const float* __restrict__ dtw,
                       const float* __restrict__ dtb,
                       const float* __restrict__ A_logs,
                       const float* __restrict__ Ds,
                       float* __restrict__ ys)
{
    const int bk = blockIdx.x;
    const int d  = threadIdx.x;
    const int k  = bk % KDIR;
    const int kd = k * D_INNER + d;

    float Wdt[DT_RANK];
#pragma unroll
    for (int r = 0; r < DT_RANK; ++r) Wdt[r] = dtw[(size_t)kd * DT_RANK + r];
    const float bias = dtb[kd];
    const float Dk   = Ds[kd];
    float Aneg[D_STATE];
#pragma unroll
    for (int n = 0; n < D_STATE; ++n)
        Aneg[n] = -expf(A_logs[(size_t)kd * D_STATE + n]);

    float hst[D_STATE];
#pragma unroll
    for (int n = 0; n < D_STATE; ++n) hst[n] = 0.f;

    const float* xsb = xs   + (size_t)bk * L * D_INNER;
    const float* xdb = xdbl + (size_t)bk * L * NPROJ_P;
    float*       yo  = ys   + (size_t)bk * L * D_INNER;

    for (int l = 0; l < L; ++l) {
        const float u = xsb[(size_t)l * D_INNER + d];
        const float* row = xdb + (size_t)l * NPROJ_P;
        float dtv = bias;
#pragma unroll
        for (int r = 0; r < DT_RANK; ++r) dtv += row[r] * Wdt[r];
        const float delta = (dtv > 20.f) ? dtv : log1pf(expf(dtv));
        const float du = delta * u;
        float acc = 0.f;
#pragma unroll
        for (int n = 0; n < D_STATE; ++n) {
            const float a = expf(delta * Aneg[n]);
            hst[n] = a * hst[n] + du * row[DT_RANK + n];
            acc += hst[n] * row[DT_RANK + D_STATE + n];
        }
        yo[(size_t)l * D_INNER + d] = acc + Dk * u;
    }
}

// ---------------------------------------------------------------------------
// K5: combine 4 directions (flip/transpose gathers) + LayerNorm(192) + gate
// with SiLU(z).  Block per (b,l), thread per d.
// ---------------------------------------------------------------------------
__global__ void k_combine_ln(const float* __restrict__ ys,
                             const float* __restrict__ z,
                             const float* __restrict__ g,
                             const float* __restrict__ beta,
                             float* __restrict__ yn)
{
    const int bl = blockIdx.x;
    const int d  = threadIdx.x;
    const int b  = bl / L;
    const int l  = bl % L;
    const int h  = l / Ww, w = l % Ww;
    const int lt = w * Hh + h;

    const float* yb = ys + (size_t)b * KDIR * L * D_INNER;
    float y = yb[((size_t)0 * L + l)            * D_INNER + d]
            + yb[((size_t)1 * L + lt)           * D_INNER + d]
            + yb[((size_t)2 * L + (L - 1 - l))  * D_INNER + d]
            + yb[((size_t)3 * L + (L - 1 - lt)) * D_INNER + d];

    // block reduction over 192 lanes (6 waves of 32)
    float s1 = y, s2 = y * y;
#pragma unroll
    for (int off = 16; off > 0; off >>= 1) {
        s1 += __shfl_xor(s1, off, 32);
        s2 += __shfl_xor(s2, off, 32);
    }
    __shared__ float ws1[6], ws2[6];
    const int wid = threadIdx.x >> 5, lane = threadIdx.x & 31;
    if (lane == 0) { ws1[wid] = s1; ws2[wid] = s2; }
    __syncthreads();
    float t1 = 0.f, t2 = 0.f;
#pragma unroll
    for (int i = 0; i < 6; ++i) { t1 += ws1[i]; t2 += ws2[i]; }
    const float mean = t1 / (float)D_INNER;
    const float var  = t2 / (float)D_INNER - mean * mean;
    const float inv  = rsqrtf(var + 1e-5f);

    float v = (y - mean) * inv * g[d] + beta[d];
    const float zv = z[(size_t)bl * D_INNER + d];
    v *= silu_f(zv);
    yn[(size_t)bl * D_INNER + d] = v;
}

// ---------------------------------------------------------------------------
// K6: out = yn @ out_proj_w^T.  M=4608, N=96, K=192.
// ---------------------------------------------------------------------------
__global__ void k_out_proj(const float* __restrict__ yn,
                           const float* __restrict__ Wo,
                           float* __restrict__ out)
{
    constexpr int NT = D_MODEL / 16;               // 6
    const int tile = blockIdx.x * (blockDim.x >> 5) + (threadIdx.x >> 5);
    const int mt = tile / NT, nt = tile % NT;

    v8f c = wmma_tile_f32(yn, D_INNER, Wo, D_INNER, mt * 16, nt * 16,
                          D_INNER, D_MODEL);

    const int lane = threadIdx.x & 31;
    const int half = lane >> 4, lr = lane & 15;
    const int n = nt * 16 + lr;
#pragma unroll
    for (int v = 0; v < 8; ++v) {
        const int m = mt * 16 + v + half * 8;
        out[(size_t)m * D_MODEL + n] = c[v];
    }
}

// ---------------------------------------------------------------------------
extern "C" void kernel_launch(void* const* d_in, const int* in_sizes, int n_in,
                              void* d_out, int out_size, void* d_ws, size_t ws_size,
                              hipStream_t stream)
{
    const float* x    = (const float*)d_in[0];
    const float* ipw  = (const float*)d_in[1];   // (384, 96)
    const float* cw   = (const float*)d_in[2];   // (192, 1, 3, 3)
    const float* cb   = (const float*)d_in[3];   // (192,)
    const float* xpw  = (const float*)d_in[4];   // (4, 38, 192)
    const float* dtw  = (const float*)d_in[5];   // (4, 192, 6)
    const float* dtb  = (const float*)d_in[6];   // (4, 192)
    const float* Alog = (const float*)d_in[7];   // (768, 16)
    const float* Dsv  = (const float*)d_in[8];   // (768,)
    const float* ong  = (const float*)d_in[9];   // (192,)
    const float* onb  = (const float*)d_in[10];  // (192,)
    const float* opw  = (const float*)d_in[11];  // (96, 192)
    float* out = (float*)d_out;

    // workspace carve-up (256B aligned regions)
    auto align_up = [](size_t v) { return (v + 255) & ~(size_t)255; };
    char* ws = (char*)d_ws;
    size_t off = 0;
    float* xc   = (float*)(ws + off); off = align_up(off + (size_t)ML * D_INNER * 4);
    float* zb   = (float*)(ws + off); off = align_up(off + (size_t)ML * D_INNER * 4);
    float* xs   = (float*)(ws + off); off = align_up(off + (size_t)Bsz * KDIR * L * D_INNER * 4);
    float* xdbl = (float*)(ws + off); off = align_up(off + (size_t)Bsz * KDIR * L * NPROJ_P * 4);
    float* ys   = (float*)(ws + off); off = align_up(off + (size_t)Bsz * KDIR * L * D_INNER * 4);
    float* yn   = (float*)(ws + off); off = align_up(off + (size_t)ML * D_INNER * 4);
    (void)ws_size;

    // K1: in_proj GEMM.  (4608/16)*(384/16)=6912 tiles, 8 waves/block -> 864
    k_in_proj<<<864, 256, 0, stream>>>(x, ipw, xc, zb);

    // K2: conv + silu + 4-direction scatter
    {
        const int n = Bsz * L * D_INNER;
        k_conv_dirs<<<(n + 255) / 256, 256, 0, stream>>>(xc, cw, cb, xs);
    }

    // K3: x_dbl GEMMs.  8*(144*3)=3456 tiles -> 432 blocks
    k_xdbl<<<432, 256, 0, stream>>>(xs, xpw, xdbl);

    // K4: selective scan, one block per (b,k)
    k_scan<<<Bsz * KDIR, D_INNER, 0, stream>>>(xs, xdbl, dtw, dtb, Alog, Dsv, ys);

    // K5: combine + layernorm + gate
    k_combine_ln<<<Bsz * L, D_INNER, 0, stream>>>(ys, zb, ong, onb, yn);

    // K6: out_proj GEMM. (4608/16)*(96/16)=1728 tiles -> 216 blocks
    k_out_proj<<<216, 256, 0, stream>>>(yn, opw, out);
}